// InstanceAdaptiveController_57226144252248
// MI455X (gfx1250) — compile-verified
//
#include <hip/hip_runtime.h>
#include <cstdint>
#include <cstddef>

#define B_      32
#define S_      4096
#define H_      2048
#define ADAPT_  32
#define RANK_   16
#define TOPK_   8
#define TEMP_   0.1f
#define NSPLIT  16
#define SCHUNK  (S_ / NSPLIT)   // 256 rows per split

typedef float v8f __attribute__((ext_vector_type(8)));
typedef float v4f __attribute__((ext_vector_type(4)));
typedef float v2f __attribute__((ext_vector_type(2)));

// ---------------------------------------------------------------------------
// JAX threefry2x32 (exact), key(seed) -> (seed>>32, seed&0xffffffff)
// ---------------------------------------------------------------------------
__device__ __forceinline__ void threefry2x32(uint32_t k0, uint32_t k1,
                                             uint32_t x0, uint32_t x1,
                                             uint32_t& o0, uint32_t& o1) {
    uint32_t ks2 = k0 ^ k1 ^ 0x1BD11BDAu;
    x0 += k0; x1 += k1;
    auto rnd = [&](uint32_t r) {
        x0 += x1; x1 = (x1 << r) | (x1 >> (32u - r)); x1 ^= x0;
    };
    rnd(13); rnd(15); rnd(26); rnd(6);
    x0 += k1;  x1 += ks2 + 1u;
    rnd(17); rnd(29); rnd(16); rnd(24);
    x0 += ks2; x1 += k0 + 2u;
    rnd(13); rnd(15); rnd(26); rnd(6);
    x0 += k0;  x1 += k1 + 3u;
    rnd(17); rnd(29); rnd(16); rnd(24);
    x0 += k1;  x1 += ks2 + 4u;
    rnd(13); rnd(15); rnd(26); rnd(6);
    x0 += ks2; x1 += k0 + 5u;
    o0 = x0; o1 = x1;
}

__device__ __forceinline__ float bits_to_u01(uint32_t bits) {
    return __uint_as_float((bits >> 9) | 0x3F800000u) - 1.0f;   // [0,1)
}

// element i of jax.random.uniform(key,(n,)) with n even (counts split n/2 | n/2)
__device__ __forceinline__ float jax_uniform(uint32_t k0, uint32_t k1,
                                             uint32_t i, uint32_t n) {
    uint32_t h = n >> 1, o0, o1;
    if (i < h) { threefry2x32(k0, k1, i, i + h, o0, o1); return bits_to_u01(o0); }
    threefry2x32(k0, k1, i - h, i, o0, o1);              return bits_to_u01(o1);
}

// ---------------------------------------------------------------------------
// Phase 1: streaming partial sums over S (bandwidth-bound, 1.07 GB read)
// grid = (H/1024, NSPLIT, B), 256 threads, ext-vector float4 per thread
// ---------------------------------------------------------------------------
__global__ void pool_partial_kernel(const float* __restrict__ hidden,
                                    float* __restrict__ partial) {
    const int h  = blockIdx.x * 1024 + threadIdx.x * 4;
    const int sp = blockIdx.y;
    const int b  = blockIdx.z;
    const v4f* src = (const v4f*)(hidden +
        ((size_t)b * S_ + (size_t)sp * SCHUNK) * H_ + h);
    v4f acc = {0.f, 0.f, 0.f, 0.f};
#pragma unroll 4
    for (int s = 0; s < SCHUNK; ++s) {
        v4f v = __builtin_nontemporal_load(src);   // NT: single-pass stream
        acc += v;
        src += H_ / 4;
    }
    v4f* dst = (v4f*)(partial + ((size_t)sp * B_ + b) * H_ + h);
    __builtin_nontemporal_store(acc, dst);
}

// Phase 2: deterministic reduce over splits -> pooled[b*H + h] (mean over S)
__global__ void pool_reduce_kernel(const float* __restrict__ partial,
                                   float* __restrict__ pooled) {
    const int idx = blockIdx.x * blockDim.x + threadIdx.x;   // 0 .. B*H-1
    float acc = 0.f;
#pragma unroll
    for (int s = 0; s < NSPLIT; ++s)
        acc += partial[(size_t)s * B_ * H_ + idx];
    pooled[idx] = acc * (1.0f / (float)S_);
}

// ---------------------------------------------------------------------------
// Phase 3: adaptation net. One block, 128 threads = 4 wave32s.
// pooled[32,2048] @ W1[2048,32] via V_WMMA_F32_16X16X4_F32 (2x2 tiles, K-loop)
// then LayerNorm -> exact GELU -> Threefry dropout -> W2 -> Gumbel top-8 ->
// straight-through mask -> EMA -> entropy.
// ---------------------------------------------------------------------------
__global__ void adapt_kernel(const float* __restrict__ pooled,
                             const float* __restrict__ W1,
                             const float* __restrict__ b1,
                             const float* __restrict__ ln_g,
                             const float* __restrict__ ln_b,
                             const float* __restrict__ W2,
                             const float* __restrict__ b2,
                             const float* __restrict__ mask_logits,
                             const float* __restrict__ mask_ema,
                             float* __restrict__ out) {
    __shared__ float h_lds[B_][ADAPT_ + 1];
    __shared__ float mask_lds[B_][RANK_];
    __shared__ float ema_lds[RANK_];

    const int tid  = threadIdx.x;
    const int wave = tid >> 5;
    const int lane = tid & 31;
    const int mt   = (wave >> 1) * 16;   // tile row base
    const int nt   = (wave & 1) * 16;    // tile col base
    const int half = lane >> 4;          // 0: K={0,1}, 1: K={2,3}
    const int l15  = lane & 15;
    const int mrow = mt + l15;           // A row for this lane
    const int ncol = nt + l15;           // B col for this lane

    // ---- 16x16 tile of h = pooled @ W1 using f32 WMMA, K in steps of 4 ----
    v8f c = {};
    const float* arow = pooled + (size_t)mrow * H_;
    for (int k0 = 0; k0 < H_; k0 += 4) {
        const int ka = k0 + 2 * half;
        v2f a, bb;
        a.x  = arow[ka];
        a.y  = arow[ka + 1];
        bb.x = W1[(size_t)ka * ADAPT_ + ncol];
        bb.y = W1[(size_t)(ka + 1) * ADAPT_ + ncol];
        c = __builtin_amdgcn_wmma_f32_16x16x4_f32(
                false, a, false, bb, (short)0, c, false, false);
    }
    // C/D layout: VGPR r -> lanes 0-15: M=mt+r, lanes 16-31: M=mt+8+r; N=nt+l15
#pragma unroll
    for (int r = 0; r < 8; ++r) {
        const int m = mt + half * 8 + r;
        const int n = nt + l15;
        h_lds[m][n] = c[r] + b1[n];
    }
    __syncthreads();

    // ---- per-batch-row tail (32 rows, one thread each) ----
    if (tid < B_) {
        const int row = tid;
        float mu = 0.f;
#pragma unroll
        for (int j = 0; j < ADAPT_; ++j) mu += h_lds[row][j];
        mu *= (1.0f / ADAPT_);
        float var = 0.f;
#pragma unroll
        for (int j = 0; j < ADAPT_; ++j) {
            const float d = h_lds[row][j] - mu;
            var += d * d;
        }
        var *= (1.0f / ADAPT_);
        const float rstd = rsqrtf(var + 1e-5f);

        float hv[ADAPT_];
#pragma unroll
        for (int j = 0; j < ADAPT_; ++j) {
            float t = (h_lds[row][j] - mu) * rstd * ln_g[j] + ln_b[j];
            // exact GELU: 0.5*x*(1+erf(x/sqrt(2)))
            t = 0.5f * t * (1.0f + erff(t * 0.70710678118654752440f));
            // dropout(0.1), jax.random.key(1) -> (0,1), 1024 elems
            const float u = jax_uniform(0u, 1u, (uint32_t)(row * ADAPT_ + j), 1024u);
            hv[j] = (u >= 0.1f) ? t * (1.0f / 0.9f) : 0.0f;
        }

        // instance_logits = hv @ W2 + b2 ; combined = + mask_logits
        float comb[RANK_];
#pragma unroll
        for (int r = 0; r < RANK_; ++r) {
            float acc = b2[r];
#pragma unroll
            for (int j = 0; j < ADAPT_; ++j) acc += hv[j] * W2[j * RANK_ + r];
            comb[r] = acc + mask_logits[r];
        }

        // gumbel noise, key(2) -> (0,2), 512 elems
        float score[RANK_];
#pragma unroll
        for (int r = 0; r < RANK_; ++r) {
            const float u = jax_uniform(0u, 2u, (uint32_t)(row * RANK_ + r), 512u);
            const float g = -logf(-logf(u + 1e-8f) + 1e-8f);
            score[r] = (comb[r] + g) / TEMP_;
        }

        // top-8 (strict >, first index wins ties like lax.top_k)
        float khot[RANK_];
        bool  used[RANK_];
#pragma unroll
        for (int r = 0; r < RANK_; ++r) { khot[r] = 0.f; used[r] = false; }
        for (int it = 0; it < TOPK_; ++it) {
            int best = -1; float bv = -3.402823466e38f;
#pragma unroll
            for (int r = 0; r < RANK_; ++r)
                if (!used[r] && score[r] > bv) { bv = score[r]; best = r; }
            used[best] = true; khot[best] = 1.0f;
        }

        // softmax(combined/TEMP) ; straight-through mask = (khot+soft)-soft
        float mx = -3.402823466e38f;
#pragma unroll
        for (int r = 0; r < RANK_; ++r) {
            const float cs = comb[r] / TEMP_;
            mx = (cs > mx) ? cs : mx;
        }
        float es[RANK_], sum = 0.f;
#pragma unroll
        for (int r = 0; r < RANK_; ++r) { es[r] = expf(comb[r] / TEMP_ - mx); sum += es[r]; }
#pragma unroll
        for (int r = 0; r < RANK_; ++r) {
            const float soft = es[r] / sum;
            const float m = (khot[r] + soft) - soft;   // fp order matters
            mask_lds[row][r] = m;
            out[row * RANK_ + r] = m;
        }
    }
    __syncthreads();

    // ---- EMA update (one thread per rank) ----
    if (tid < RANK_) {
        const int r = tid;
        float s = 0.f;
#pragma unroll
        for (int row = 0; row < B_; ++row) s += mask_lds[row][r];
        const float ema = 0.99f * mask_ema[r] + 0.01f * (s / (float)B_);
        ema_lds[r] = ema;
        out[B_ * RANK_ + r] = ema;
    }
    __syncthreads();

    // ---- entropy scalar ----
    if (tid == 0) {
        float ent = 0.f;
#pragma unroll
        for (int r = 0; r < RANK_; ++r)
            ent -= ema_lds[r] * logf(ema_lds[r] + 1e-8f);
        out[B_ * RANK_ + RANK_] = ent;
    }
}

// ---------------------------------------------------------------------------
extern "C" void kernel_launch(void* const* d_in, const int* in_sizes, int n_in,
                              void* d_out, int out_size, void* d_ws, size_t ws_size,
                              hipStream_t stream) {
    const float* hidden      = (const float*)d_in[0];
    const float* W1          = (const float*)d_in[1];
    const float* b1          = (const float*)d_in[2];
    const float* ln_g        = (const float*)d_in[3];
    const float* ln_b        = (const float*)d_in[4];
    const float* W2          = (const float*)d_in[5];
    const float* b2          = (const float*)d_in[6];
    const float* mask_logits = (const float*)d_in[7];
    const float* mask_ema    = (const float*)d_in[8];
    float* out = (float*)d_out;

    float* partial = (float*)d_ws;                           // NSPLIT*B*H floats
    float* pooled  = partial + (size_t)NSPLIT * B_ * H_;     // B*H floats

    dim3 gp(H_ / 1024, NSPLIT, B_);                          // 1024 blocks
    pool_partial_kernel<<<gp, 256, 0, stream>>>(hidden, partial);
    pool_reduce_kernel<<<(B_ * H_) / 256, 256, 0, stream>>>(partial, pooled);
    adapt_kernel<<<1, 128, 0, stream>>>(pooled, W1, b1, ln_g, ln_b,
                                        W2, b2, mask_logits, mask_ema, out);
}